// ChiSquareLoss_69166153335036
// MI455X (gfx1250) — compile-verified
//
#include <hip/hip_runtime.h>
#include <cstdint>

// ---------------------------------------------------------------------------
// ChiSquareLoss for MI455X (gfx1250).
// 16384x2048 fp32 = 128 MB read once => ~5.5us floor @ 23.3 TB/s.
// One wave32 per row; stage the 8KB row in LDS via global_load_async_to_lds
// (ASYNCcnt path), then two LDS passes. Binning counters are split across
// the VALU pipe (4 per-lane counters) and the SALU pipe (5 ballot+s_bcnt
// wave-uniform counters). The binning loop is only partially unrolled so the
// scheduler does not hoist ballot SGPR results far from their s_bcnt
// consumers (round 2 fully-unrolled version caused SGPR spills).
// ---------------------------------------------------------------------------

#define BINS_N 10
constexpr int D_DIM        = 2048;   // features per row (fixed by reference)
constexpr int WAVES_PER_WG = 8;     // 256 threads / block, 8 rows / block
constexpr int LANES        = 32;    // wave32
constexpr int CHUNKS       = D_DIM / (LANES * 4);   // 16 float4 chunks per lane

__global__ __launch_bounds__(256) void row_chi2_kernel(
    const float* __restrict__ emb, float* __restrict__ rowchi,
    int nrows, float expected) {
  __shared__ float tile[WAVES_PER_WG][D_DIM];   // 64 KB

  const int lane = threadIdx.x & (LANES - 1);
  const int wave = threadIdx.x >> 5;
  const int row  = blockIdx.x * WAVES_PER_WG + wave;
  if (row >= nrows) return;   // wave-uniform branch

  // LDS byte offset for this lane's first 16B chunk (low 32 bits of the
  // generic LDS address are the workgroup-relative LDS offset).
  unsigned lds_off = (unsigned)(uintptr_t)&tile[wave][lane * 4];
  unsigned long long gaddr =
      (unsigned long long)(uintptr_t)(emb + (size_t)row * D_DIM + (size_t)lane * 4);

  // 16 async 128b copies per wave: 32 lanes x 16B = 512B per instruction.
  // The same immediate offset applies to both the global and LDS address.
#define ROW_ASYNC_LD(OFF)                                                   \
  asm volatile("global_load_async_to_lds_b128 %0, %1, off offset:" #OFF    \
               :: "v"(lds_off), "v"(gaddr) : "memory")
  ROW_ASYNC_LD(0);    ROW_ASYNC_LD(512);  ROW_ASYNC_LD(1024); ROW_ASYNC_LD(1536);
  ROW_ASYNC_LD(2048); ROW_ASYNC_LD(2560); ROW_ASYNC_LD(3072); ROW_ASYNC_LD(3584);
  ROW_ASYNC_LD(4096); ROW_ASYNC_LD(4608); ROW_ASYNC_LD(5120); ROW_ASYNC_LD(5632);
  ROW_ASYNC_LD(6144); ROW_ASYNC_LD(6656); ROW_ASYNC_LD(7168); ROW_ASYNC_LD(7680);
#undef ROW_ASYNC_LD

  // Only this wave reads the tile it wrote -> per-wave async wait suffices.
  asm volatile("s_wait_asynccnt 0x0" ::: "memory");

  const float4* t4 = reinterpret_cast<const float4*>(&tile[wave][0]);

  // ---- pass 1: per-row min / max --------------------------------------
  float mn = 3.402823466e+38f;
  float mx = -3.402823466e+38f;
#pragma unroll
  for (int i = 0; i < CHUNKS; ++i) {
    float4 v = t4[lane + LANES * i];
    mn = fminf(mn, fminf(fminf(v.x, v.y), fminf(v.z, v.w)));
    mx = fmaxf(mx, fmaxf(fmaxf(v.x, v.y), fmaxf(v.z, v.w)));
  }
#pragma unroll
  for (int off = 16; off > 0; off >>= 1) {
    mn = fminf(mn, __shfl_xor(mn, off, 32));
    mx = fmaxf(mx, __shfl_xor(mx, off, 32));
  }

  // interior boundaries b_k = mn + (mx-mn) * k/10, k = 1..9 (fp32, as ref)
  float bnd[BINS_N - 1];
#pragma unroll
  for (int k = 1; k < BINS_N; ++k)
    bnd[k - 1] = mn + (mx - mn) * ((float)k / (float)BINS_N);

  // ---- pass 2: G_k = #{x > b_k} ---------------------------------------
  // searchsorted(side='left') => digitized = #{b < x}; histogram recovered
  // as observed[j] = G_j - G_{j+1}, G_0 = D, G_10 = 0.
  // Bins 0..3: per-lane VGPR counters (v_cmp + carry-add, VALU pipe).
  // Bins 4..8: wave-uniform ballot+popcount counters (s_bcnt, SALU pipe).
  int c0 = 0, c1 = 0, c2 = 0, c3 = 0;                // per-lane (VALU)
  int s4 = 0, s5 = 0, s6 = 0, s7 = 0, s8 = 0;        // wave-uniform (SALU)

#pragma unroll 4
  for (int i = 0; i < CHUNKS; ++i) {
    float4 v = t4[lane + LANES * i];
#pragma unroll
    for (int e = 0; e < 4; ++e) {
      const float x = (e == 0) ? v.x : (e == 1) ? v.y : (e == 2) ? v.z : v.w;
      c0 += (bnd[0] < x) ? 1 : 0;
      c1 += (bnd[1] < x) ? 1 : 0;
      c2 += (bnd[2] < x) ? 1 : 0;
      c3 += (bnd[3] < x) ? 1 : 0;
      s4 += __builtin_popcount(__builtin_amdgcn_ballot_w32(bnd[4] < x));
      s5 += __builtin_popcount(__builtin_amdgcn_ballot_w32(bnd[5] < x));
      s6 += __builtin_popcount(__builtin_amdgcn_ballot_w32(bnd[6] < x));
      s7 += __builtin_popcount(__builtin_amdgcn_ballot_w32(bnd[7] < x));
      s8 += __builtin_popcount(__builtin_amdgcn_ballot_w32(bnd[8] < x));
    }
  }

  // cross-lane totals for the four per-lane counters
#pragma unroll
  for (int off = 16; off > 0; off >>= 1) {
    c0 += __shfl_xor(c0, off, 32);
    c1 += __shfl_xor(c1, off, 32);
    c2 += __shfl_xor(c2, off, 32);
    c3 += __shfl_xor(c3, off, 32);
  }

  if (lane == 0) {
    int g[BINS_N - 1] = {c0, c1, c2, c3, s4, s5, s6, s7, s8};
    const float denom = expected + 1e-8f;   // expected = nrows / BINS (as ref)
    float chi = 0.0f;
    int prev = D_DIM;                       // G_0
#pragma unroll
    for (int j = 0; j < BINS_N; ++j) {
      int nxt   = (j < BINS_N - 1) ? g[j] : 0;   // G_{j+1}
      float obs = (float)(prev - nxt);
      float d   = obs - expected;
      chi += d * d / denom;
      prev = nxt;
    }
    rowchi[row] = chi;
  }
}

// ---- final mean over rows (deterministic tree reduction, no atomics) ----
__global__ __launch_bounds__(256) void mean_reduce_kernel(
    const float* __restrict__ rowchi, float* __restrict__ out, int n) {
  __shared__ float smem[256];
  float acc = 0.0f;
  for (int i = (int)threadIdx.x; i < n; i += 256) acc += rowchi[i];
  smem[threadIdx.x] = acc;
  __syncthreads();
#pragma unroll
  for (int off = 128; off > 0; off >>= 1) {
    if ((int)threadIdx.x < off) smem[threadIdx.x] += smem[threadIdx.x + off];
    __syncthreads();
  }
  if (threadIdx.x == 0) out[0] = smem[0] / (float)n;
}

extern "C" void kernel_launch(void* const* d_in, const int* in_sizes, int n_in,
                              void* d_out, int out_size, void* d_ws, size_t ws_size,
                              hipStream_t stream) {
  const float* emb = (const float*)d_in[0];
  float* rowchi    = (float*)d_ws;          // nrows floats of scratch
  float* out       = (float*)d_out;

  const int nrows      = in_sizes[0] / D_DIM;          // 16384
  const float expected = (float)nrows / (float)BINS_N; // 1638.4 (ref uses B/BINS)
  const int blocks     = (nrows + WAVES_PER_WG - 1) / WAVES_PER_WG;

  row_chi2_kernel<<<blocks, 256, 0, stream>>>(emb, rowchi, nrows, expected);
  mean_reduce_kernel<<<1, 256, 0, stream>>>(rowchi, out, nrows);
}